// GNNEncoder_76802605187487
// MI455X (gfx1250) — compile-verified
//
#include <hip/hip_runtime.h>

typedef float v2f __attribute__((ext_vector_type(2)));
typedef float v8f __attribute__((ext_vector_type(8)));

#define D64 64
#define BN_EPS 1e-5f
#define LEAKY 0.01f

// Native hardware FP32 atomic add (global_atomic_add_f32), no CAS fallback.
__device__ __forceinline__ void atomic_add_f32(float* p, float v) {
  unsafeAtomicAdd(p, v);
}

// ---------------- utility fills ----------------
__global__ void fill_f32(float* __restrict__ p, float v, long long n) {
  long long gid = (long long)blockIdx.x * blockDim.x + threadIdx.x;
  if (gid < n) p[gid] = v;
}

// ---------------- degree / norm ----------------
__global__ void deg_accum(const int* __restrict__ dst, float* __restrict__ deg, int E) {
  int e = blockIdx.x * blockDim.x + threadIdx.x;
  if (e < E) atomic_add_f32(&deg[dst[e]], 1.0f);
}

__global__ void rsqrt_inplace(float* __restrict__ p, int n) {
  int i = blockIdx.x * blockDim.x + threadIdx.x;
  if (i < n) p[i] = rsqrtf(p[i]);  // deg >= 1 always (self-loops)
}

// ---------------- dense GEMM: Y[n,64] = X[n,64] @ W[64,64] via WMMA f32 16x16x4 ----
// Block = 128 threads = 4 waves; each wave computes a 16-row x 64-col tile.
__global__ void gemm64_wmma(const float* __restrict__ X, const float* __restrict__ W,
                            float* __restrict__ Y, int n) {
  const int wave = threadIdx.x >> 5;
  const int lane = threadIdx.x & 31;
  const int row0 = (blockIdx.x * 4 + wave) * 16;
  if (row0 >= n) return;                 // wave-uniform: EXEC stays all-1 for WMMA
  const int m    = lane & 15;            // M index (A) / N index (B,C)
  const int half = lane >> 4;            // 0: K={0,1}  1: K={2,3}
  const int koff = half * 2;

  int rload = row0 + m;
  if (rload >= n) rload = n - 1;         // clamp tail loads (stores are guarded)
  const float* xr = X + (size_t)rload * D64;

  v8f c0 = {}; v8f c1 = {}; v8f c2 = {}; v8f c3 = {};
#pragma unroll
  for (int kk = 0; kk < D64; kk += 4) {
    v2f a;
    a.x = xr[kk + koff];
    a.y = xr[kk + koff + 1];
    const float* wk = W + (size_t)(kk + koff) * D64 + m;
    v2f b0, b1, b2, b3;
    b0.x = wk[0];   b0.y = wk[D64];
    b1.x = wk[16];  b1.y = wk[D64 + 16];
    b2.x = wk[32];  b2.y = wk[D64 + 32];
    b3.x = wk[48];  b3.y = wk[D64 + 48];
    c0 = __builtin_amdgcn_wmma_f32_16x16x4_f32(false, a, false, b0, (short)0, c0, false, false);
    c1 = __builtin_amdgcn_wmma_f32_16x16x4_f32(false, a, false, b1, (short)0, c1, false, false);
    c2 = __builtin_amdgcn_wmma_f32_16x16x4_f32(false, a, false, b2, (short)0, c2, false, false);
    c3 = __builtin_amdgcn_wmma_f32_16x16x4_f32(false, a, false, b3, (short)0, c3, false, false);
  }
  // C/D layout: VGPR r -> lanes 0-15: M=r, lanes 16-31: M=r+8; N = lane&15
#pragma unroll
  for (int r = 0; r < 8; ++r) {
    int row = row0 + r + half * 8;
    if (row < n) {
      float* yr = Y + (size_t)row * D64 + m;
      yr[0]  = c0[r];
      yr[16] = c1[r];
      yr[32] = c2[r];
      yr[48] = c3[r];
    }
  }
}

// ---------------- edge scatter: AGG[dst] += H[src] * dinv[src]*dinv[dst] ----------
// One thread per (edge, column): src/dst loads are wave-uniform, row reads coalesced.
// Feature matrix (25.6 MB) is L2-resident (192 MB), so gathers/atomics run at L2 speed.
__global__ void scatter_edges(const float* __restrict__ H, const int* __restrict__ src,
                              const int* __restrict__ dst, const float* __restrict__ dinv,
                              float* __restrict__ AGG, long long total) {
  long long gid = (long long)blockIdx.x * blockDim.x + threadIdx.x;
  if (gid >= total) return;
  int e = (int)(gid >> 6);
  int c = (int)(gid & 63);
  int s = src[e], d = dst[e];
  float nrm = dinv[s] * dinv[d];
  float val = H[(size_t)s * D64 + c] * nrm;
  atomic_add_f32(&AGG[(size_t)d * D64 + c], val);
}

// ---------------- self-loop + bias: AGG[i,c] += H[i,c]*dinv[i]^2 + b[c] -----------
__global__ void finish_layer(float* __restrict__ AGG, const float* __restrict__ H,
                             const float* __restrict__ dinv, const float* __restrict__ bias,
                             long long total) {
  long long gid = (long long)blockIdx.x * blockDim.x + threadIdx.x;
  if (gid >= total) return;
  int i = (int)(gid >> 6);
  int c = (int)(gid & 63);
  float di = dinv[i];
  AGG[gid] = AGG[gid] + H[gid] * di * di + bias[c];
}

// ---------------- BatchNorm stats: per-column sum & sumsq over N rows -------------
__global__ void bn_stats(const float* __restrict__ H, float* __restrict__ sums,
                         float* __restrict__ sumsq, int n) {
  __shared__ float s1[256];
  __shared__ float s2[256];
  int c    = threadIdx.x & 63;
  int rgrp = threadIdx.x >> 6;  // 0..3
  float a = 0.f, b = 0.f;
  for (int i = blockIdx.x * 4 + rgrp; i < n; i += gridDim.x * 4) {
    float v = H[(size_t)i * D64 + c];
    a += v;
    b += v * v;
  }
  s1[threadIdx.x] = a;
  s2[threadIdx.x] = b;
  __syncthreads();
  if (threadIdx.x < 64) {
    a = s1[c] + s1[c + 64] + s1[c + 128] + s1[c + 192];
    b = s2[c] + s2[c + 64] + s2[c + 128] + s2[c + 192];
    atomic_add_f32(&sums[c],  a);
    atomic_add_f32(&sumsq[c], b);
  }
}

// ---------------- BN apply + LeakyReLU --------------------------------------------
__global__ void bn_apply(const float* __restrict__ Hin, float* __restrict__ Hout,
                         const float* __restrict__ sums, const float* __restrict__ sumsq,
                         const float* __restrict__ gamma, const float* __restrict__ beta,
                         int n, long long total) {
  long long gid = (long long)blockIdx.x * blockDim.x + threadIdx.x;
  if (gid >= total) return;
  int c = (int)(gid & 63);
  float invn = 1.0f / (float)n;
  float mu   = sums[c] * invn;
  float var  = sumsq[c] * invn - mu * mu;     // biased variance
  float v = gamma[c] * (Hin[gid] - mu) * rsqrtf(var + BN_EPS) + beta[c];
  Hout[gid] = (v >= 0.f) ? v : LEAKY * v;
}

// ---------------- host-side orchestration -----------------------------------------
extern "C" void kernel_launch(void* const* d_in, const int* in_sizes, int n_in,
                              void* d_out, int out_size, void* d_ws, size_t ws_size,
                              hipStream_t stream) {
  const float* x     = (const float*)d_in[0];
  const int*   ei    = (const int*)  d_in[1];
  const float* W1    = (const float*)d_in[2];
  const float* b1    = (const float*)d_in[3];
  const float* gamma = (const float*)d_in[4];
  const float* beta  = (const float*)d_in[5];
  const float* W2    = (const float*)d_in[6];
  const float* b2    = (const float*)d_in[7];
  float* out = (float*)d_out;

  const int N = in_sizes[0] / D64;
  const int E = in_sizes[1] / 2;
  const int* src = ei;
  const int* dst = ei + E;

  // workspace layout
  char* ws = (char*)d_ws;
  size_t off = 0;
  float* deg = (float*)(ws + off);  off += (((size_t)N * 4) + 255) & ~(size_t)255;
  float* sums  = (float*)(ws + off); off += 256;  // 64 floats (+pad)
  float* sumsq = (float*)(ws + off); off += 256;
  float* hA = (float*)(ws + off);   off += (size_t)N * D64 * 4;
  float* hB = (float*)(ws + off);   off += (size_t)N * D64 * 4;
  (void)ws_size;

  const long long ND = (long long)N * D64;
  const long long ED = (long long)E * D64;
  const int T = 256;
  dim3 bT(T);
  auto blk = [](long long n, int t) { return dim3((unsigned)((n + t - 1) / t)); };

  // zero-init (atomics accumulate -> must re-zero every call)
  fill_f32<<<blk(N, T), bT, 0, stream>>>(deg, 1.0f, N);          // self-loop degree
  fill_f32<<<blk(128, T), bT, 0, stream>>>(sums, 0.0f, 128);     // sums + sumsq
  fill_f32<<<blk(ND, T), bT, 0, stream>>>(hB, 0.0f, ND);
  fill_f32<<<blk(ND, T), bT, 0, stream>>>(out, 0.0f, ND);

  // symmetric normalization
  deg_accum<<<blk(E, T), bT, 0, stream>>>(dst, deg, E);
  rsqrt_inplace<<<blk(N, T), bT, 0, stream>>>(deg, N);           // deg -> dinv
  const float* dinv = deg;

  // layer 1
  gemm64_wmma<<<dim3((N + 63) / 64), dim3(128), 0, stream>>>(x, W1, hA, N);
  scatter_edges<<<blk(ED, T), bT, 0, stream>>>(hA, src, dst, dinv, hB, ED);
  finish_layer<<<blk(ND, T), bT, 0, stream>>>(hB, hA, dinv, b1, ND);

  // BN + LeakyReLU
  bn_stats<<<dim3(1024), bT, 0, stream>>>(hB, sums, sumsq, N);
  bn_apply<<<blk(ND, T), bT, 0, stream>>>(hB, hA, sums, sumsq, gamma, beta, N, ND);

  // layer 2 -> d_out
  gemm64_wmma<<<dim3((N + 63) / 64), dim3(128), 0, stream>>>(hA, W2, hB, N);
  scatter_edges<<<blk(ED, T), bT, 0, stream>>>(hB, src, dst, dinv, out, ED);
  finish_layer<<<blk(ND, T), bT, 0, stream>>>(out, hB, dinv, b2, ND);
}